// KeyedAvgpool2d_2413771620972
// MI455X (gfx1250) — compile-verified
//
#include <hip/hip_runtime.h>

// 3x3 stride-2 VALID average pooling, f32, via CDNA5 Tensor Data Mover:
// one TENSOR_LOAD_TO_LDS per workgroup stages a 257x33 input tile in LDS,
// 8 waves compute a 128x16 output tile from LDS. Memory-bound op: WMMA is
// deliberately NOT used (AI ~0.86 FLOP/B; HBM floor ~29us at 23.3 TB/s).

typedef unsigned int u32x4 __attribute__((ext_vector_type(4)));
typedef int          i32x8 __attribute__((ext_vector_type(8)));
typedef int          i32x4 __attribute__((ext_vector_type(4)));

#define IN_H   512
#define IN_W   512
#define OUT_H  255
#define OUT_W  255
#define TW     128            // output tile width
#define TH     16             // output tile height
#define TILE_W (2 * TW + 1)   // 257 input cols
#define TILE_H (2 * TH + 1)   // 33 input rows

__global__ __launch_bounds__(256)
void avgpool3x3s2_tdm_kernel(const float* __restrict__ x, float* __restrict__ y) {
    __shared__ float tile[TILE_H * TILE_W];   // 33924 bytes

    const int bc  = blockIdx.z;          // fused (batch, channel) plane, 0..511
    const int wo0 = blockIdx.x * TW;     // output tile origin
    const int ho0 = blockIdx.y * TH;

    // Clamp the input tile origin so the TDM transfer is always in-bounds;
    // edge tiles read LDS at a shifted offset instead.
    int w0 = 2 * wo0; if (w0 > IN_W - TILE_W) w0 = IN_W - TILE_W;
    int h0 = 2 * ho0; if (h0 > IN_H - TILE_H) h0 = IN_H - TILE_H;
    const int dw = 2 * wo0 - w0;         // 0 or 1
    const int dh = 2 * ho0 - h0;         // 0 or 1

    // ---- Wave 0 issues the Tensor Data Mover 2D tile load (global -> LDS) ----
    if (threadIdx.y == 0 && threadIdx.x < 32) {
        const float* gp = x + ((size_t)bc * IN_H + h0) * IN_W + w0;
        unsigned long long ga = (unsigned long long)gp;
        unsigned lds_addr = (unsigned)(unsigned long long)(&tile[0]);

        // D# group 0 (128b): [1:0]=count=1, [63:32]=lds_addr,
        // [120:64]=global_addr, [127:126]=type=2
        u32x4 g0 = {
            1u,
            lds_addr,
            (unsigned)(ga & 0xFFFFFFFFull),
            (unsigned)((ga >> 32) & 0x01FFFFFFull) | 0x80000000u
        };

        // D# group 1 (256b):
        //  d0: workgroup_mask=0, data_size=2 (4 bytes), no pad/iterate/barrier
        //  d1[31:16] = tensor_dim0[15:0]       (512)
        //  d2[15:0]  = tensor_dim0[31:16], d2[31:16] = tensor_dim1[15:0] (512)
        //  d3[15:0]  = tensor_dim1[31:16], d3[31:16] = tile_dim0 (257)
        //  d4[15:0]  = tile_dim1 (33), d4[31:16] = tile_dim2 (0)
        //  d5        = tensor_dim0_stride[31:0] (512)
        //  d6,d7     = stride hi bits / tensor_dim1_stride (unused, 0)
        i32x8 g1 = {
            (int)0x00020000,
            (int)(((unsigned)IN_W & 0xFFFFu) << 16),
            (int)((((unsigned)IN_H & 0xFFFFu) << 16) | ((unsigned)IN_W >> 16)),
            (int)((((unsigned)TILE_W) << 16)        | ((unsigned)IN_H >> 16)),
            (int)(TILE_H),
            (int)(IN_W),
            0,
            0
        };
        i32x4 gz4 = { 0, 0, 0, 0 };              // groups 2/3 unused (2D tile)
        i32x8 gz8 = { 0, 0, 0, 0, 0, 0, 0, 0 };  // trailing group (clang-23 6-arg form)

        __builtin_amdgcn_tensor_load_to_lds(g0, g1, gz4, gz4, gz8, 0);
        __builtin_amdgcn_s_wait_tensorcnt(0);   // s_wait_tensorcnt 0
    }
    __syncthreads();

    // ---- Compute 128x16 output tile from LDS ----
    const int tx = threadIdx.x;          // 0..127 -> output column
    const int ty = threadIdx.y;          // 0..1
    const int wo = wo0 + tx;
    const float inv9 = 1.0f / 9.0f;

    if (wo < OUT_W) {
        const int c = 2 * tx + dw;
        #pragma unroll
        for (int k = 0; k < TH / 2; ++k) {
            const int hoL = ty + 2 * k;
            const int ho  = ho0 + hoL;
            if (ho < OUT_H) {
                const float* p = &tile[(2 * hoL + dh) * TILE_W + c];
                float s = p[0]            + p[1]            + p[2]
                        + p[TILE_W]       + p[TILE_W + 1]   + p[TILE_W + 2]
                        + p[2 * TILE_W]   + p[2 * TILE_W+1] + p[2 * TILE_W+2];
                y[((size_t)bc * OUT_H + ho) * OUT_W + wo] = s * inv9;
            }
        }
    }
}

extern "C" void kernel_launch(void* const* d_in, const int* in_sizes, int n_in,
                              void* d_out, int out_size, void* d_ws, size_t ws_size,
                              hipStream_t stream) {
    const float* x = (const float*)d_in[0];   // (8, 64, 512, 512) f32
    float* y = (float*)d_out;                 // (8, 64, 255, 255) f32

    dim3 block(128, 2, 1);                    // 256 threads = 8 wave32
    dim3 grid((OUT_W + TW - 1) / TW,          // 2
              (OUT_H + TH - 1) / TH,          // 16
              8 * 64);                        // 512 (b,c) planes
    avgpool3x3s2_tdm_kernel<<<grid, block, 0, stream>>>(x, y);
}